// AttentionPooling_15960098472034
// MI455X (gfx1250) — compile-verified
//
#include <hip/hip_runtime.h>
#include <math.h>

typedef __attribute__((ext_vector_type(2))) float v2f;
typedef __attribute__((ext_vector_type(8))) float v8f;

#define BATCH 8
#define NTOK  65536
#define DIM   128
#define HID   64
#define NS    8192          // NTOK * 0.125
#define CHUNK 2048          // elements per LDS bitonic chunk
#define W1_STRIDE 130       // 128 + 2 pad -> no LDS bank conflicts

// ---------------------------------------------------------------------------
// Stage 1: scoring MLP with f32 WMMA.  One wave = one 16-row tile of x.
// h(16x64) = tanh(x(16x128) @ W1^T(128x64) + b1); s = sigmoid(h . V + vb).
// Emits packed sort key: (bits(s)<<32) | (0xFFFFFFFF - token_index).
// ---------------------------------------------------------------------------
__global__ __launch_bounds__(256) void score_kernel(
    const float* __restrict__ x, const float* __restrict__ W1w,
    const float* __restrict__ W1b, const float* __restrict__ Vw,
    const float* __restrict__ Vb, unsigned long long* __restrict__ keys) {
  __shared__ float w1s[HID * W1_STRIDE];
  __shared__ float b1s[HID];
  __shared__ float vws[HID];
  __shared__ float vbs;

  for (int i = threadIdx.x; i < HID * DIM; i += 256) {
    int h = i >> 7, d = i & (DIM - 1);
    w1s[h * W1_STRIDE + d] = W1w[i];
  }
  if (threadIdx.x < HID) {
    b1s[threadIdx.x] = W1b[threadIdx.x];
    vws[threadIdx.x] = Vw[threadIdx.x];
  }
  if (threadIdx.x == 0) vbs = Vb[0];
  __syncthreads();

  const int wave = threadIdx.x >> 5;
  const int lane = threadIdx.x & 31;
  const int half = lane >> 4;       // 0: K pair {0,1}; 1: K pair {2,3}
  const int l    = lane & 15;
  const int tile = blockIdx.x * 8 + wave;   // 16-row tile over flat B*N rows
  const int r0   = tile * 16;               // tiles never straddle a batch

  v8f acc[4] = {};                          // 4 tiles of 16 hid cols each

  const float* arow = x + (size_t)(r0 + l) * DIM + 2 * half;
  for (int k0 = 0; k0 < DIM; k0 += 4) {
    v2f a = *(const v2f*)(arow + k0);       // A: M=l, K={k0+2h, k0+2h+1}
#pragma unroll
    for (int th = 0; th < 4; ++th) {
      const int hcol = th * 16 + l;         // B col = hid index
      v2f bm = *(const v2f*)(&w1s[hcol * W1_STRIDE + k0 + 2 * half]);
      acc[th] = __builtin_amdgcn_wmma_f32_16x16x4_f32(
          false, a, false, bm, (short)0, acc[th], false, false);
    }
  }

  // C layout: VGPR r, lanes 0-15 -> (M=r, N=l); lanes 16-31 -> (M=r+8, N=l).
  const float vb = vbs;
#pragma unroll
  for (int r = 0; r < 8; ++r) {
    float pr = 0.f;
#pragma unroll
    for (int th = 0; th < 4; ++th) {
      const int hidx = th * 16 + l;
      pr += tanhf(acc[th][r] + b1s[hidx]) * vws[hidx];
    }
    // reduce over the 16 lanes of this half-wave
    pr += __shfl_xor(pr, 1);
    pr += __shfl_xor(pr, 2);
    pr += __shfl_xor(pr, 4);
    pr += __shfl_xor(pr, 8);
    if (l == 0) {
      const int m = r + half * 8;
      const int g = r0 + m;                       // flat row in [0, B*N)
      const unsigned n = (unsigned)g & (NTOK - 1);
      const float sv = 1.f / (1.f + expf(-(pr + vb)));
      const unsigned kb = __float_as_uint(sv);    // s in (0,1) -> monotonic
      keys[g] = ((unsigned long long)kb << 32) |
                (unsigned long long)(0xFFFFFFFFu - n);
    }
  }
}

// ---------------------------------------------------------------------------
// Stage 2: per-batch descending bitonic sort of 64-bit keys.
// Direction uses the batch-local index (gi & (NTOK-1)).
// ---------------------------------------------------------------------------
__device__ __forceinline__ void cmp_swap(unsigned long long& a,
                                         unsigned long long& b, bool desc) {
  if ((a < b) == desc) { unsigned long long t = a; a = b; b = t; }
}

// all stages k = 2..CHUNK entirely inside one LDS chunk
__global__ __launch_bounds__(256) void bitonic_first(
    unsigned long long* __restrict__ keys) {
  __shared__ unsigned long long sk[CHUNK];
  const size_t base = (size_t)blockIdx.x * CHUNK;
  for (int i = threadIdx.x; i < CHUNK; i += 256) sk[i] = keys[base + i];
  __syncthreads();
  for (int k = 2; k <= CHUNK; k <<= 1) {
    for (int j = k >> 1; j >= 1; j >>= 1) {
      for (int p = threadIdx.x; p < CHUNK / 2; p += 256) {
        const int low = p & (j - 1);
        const int i = ((p ^ low) << 1) | low;
        const int gi = (int)((base + i) & (NTOK - 1));
        unsigned long long a = sk[i], b = sk[i + j];
        cmp_swap(a, b, (gi & k) == 0);
        sk[i] = a; sk[i + j] = b;
      }
      __syncthreads();
    }
  }
  for (int i = threadIdx.x; i < CHUNK; i += 256) keys[base + i] = sk[i];
}

// one global compare-exchange pass for large j (>= CHUNK)
__global__ __launch_bounds__(256) void bitonic_global(
    unsigned long long* __restrict__ keys, int j, int k) {
  const int t = blockIdx.x * 256 + threadIdx.x;   // pair id, < B*N/2
  const int low = t & (j - 1);
  const int i = ((t ^ low) << 1) | low;
  const int gi = i & (NTOK - 1);
  unsigned long long a = keys[i], b = keys[i + j];
  cmp_swap(a, b, (gi & k) == 0);
  keys[i] = a; keys[i + j] = b;
}

// all remaining j = CHUNK/2 .. 1 of stage k, fused in LDS
__global__ __launch_bounds__(256) void bitonic_inner(
    unsigned long long* __restrict__ keys, int k) {
  __shared__ unsigned long long sk[CHUNK];
  const size_t base = (size_t)blockIdx.x * CHUNK;
  for (int i = threadIdx.x; i < CHUNK; i += 256) sk[i] = keys[base + i];
  __syncthreads();
  for (int j = CHUNK / 2; j >= 1; j >>= 1) {
    for (int p = threadIdx.x; p < CHUNK / 2; p += 256) {
      const int low = p & (j - 1);
      const int i = ((p ^ low) << 1) | low;
      const int gi = (int)((base + i) & (NTOK - 1));
      unsigned long long a = sk[i], b = sk[i + j];
      cmp_swap(a, b, (gi & k) == 0);
      sk[i] = a; sk[i + j] = b;
    }
    __syncthreads();
  }
  for (int i = threadIdx.x; i < CHUNK; i += 256) keys[base + i] = sk[i];
}

// ---------------------------------------------------------------------------
// Stage 3: gather top NS rows per batch.  One wave per output row, float4/lane.
// d_out layout: x_g (B*NS*D) | x_v_g (B*NS*D) | x_s_g (B*NS)
// ---------------------------------------------------------------------------
__global__ __launch_bounds__(32) void gather_kernel(
    const float* __restrict__ x, const float* __restrict__ xv,
    const unsigned long long* __restrict__ keys, float* __restrict__ out) {
  const int row = blockIdx.x;                 // 0 .. B*NS-1
  const int b = row / NS;
  const int m = row - b * NS;
  const unsigned long long c = keys[(size_t)b * NTOK + m];
  const unsigned n = 0xFFFFFFFFu - (unsigned)(c & 0xFFFFFFFFull);
  const float s = __uint_as_float((unsigned)(c >> 32));

  const float4* src  = (const float4*)(x  + ((size_t)b * NTOK + n) * DIM);
  const float4* srcv = (const float4*)(xv + ((size_t)b * NTOK + n) * DIM);
  float4* dx = (float4*)(out + (size_t)row * DIM);
  float4* dv = (float4*)(out + (size_t)BATCH * NS * DIM + (size_t)row * DIM);

  const int t = threadIdx.x;                  // 32 lanes * float4 = 128 floats
  dx[t] = src[t];
  dv[t] = srcv[t];
  if (t == 0) out[(size_t)2 * BATCH * NS * DIM + row] = s;
}

// ---------------------------------------------------------------------------
extern "C" void kernel_launch(void* const* d_in, const int* in_sizes, int n_in,
                              void* d_out, int out_size, void* d_ws, size_t ws_size,
                              hipStream_t stream) {
  const float* x   = (const float*)d_in[0];
  const float* xv  = (const float*)d_in[1];
  const float* W1w = (const float*)d_in[2];
  const float* W1b = (const float*)d_in[3];
  const float* Vw  = (const float*)d_in[4];
  const float* Vb  = (const float*)d_in[5];
  float* out = (float*)d_out;
  unsigned long long* keys = (unsigned long long*)d_ws;  // B*N*8 = 4 MB

  // 1) scores -> packed keys.  32768 tiles, 8 waves/block.
  score_kernel<<<(BATCH * NTOK / 16) / 8, 256, 0, stream>>>(x, W1w, W1b, Vw, Vb, keys);

  // 2) per-batch descending bitonic sort (LDS-fused small-j stages).
  const int nchunks = BATCH * NTOK / CHUNK;               // 256
  bitonic_first<<<nchunks, 256, 0, stream>>>(keys);
  for (int k = 2 * CHUNK; k <= NTOK; k <<= 1) {
    for (int j = k >> 1; j >= CHUNK; j >>= 1)
      bitonic_global<<<(BATCH * NTOK / 2) / 256, 256, 0, stream>>>(keys, j, k);
    bitonic_inner<<<nchunks, 256, 0, stream>>>(keys, k);
  }

  // 3) gather top-NS rows (keys[b*N .. b*N+NS) are the sorted top).
  gather_kernel<<<BATCH * NS, 32, 0, stream>>>(x, xv, keys, out);
}